// SSIM_33449205301581
// MI455X (gfx1250) — compile-verified
//
#include <hip/hip_runtime.h>

// ---------------------------------------------------------------------------
// SSIM on MI455X (gfx1250, wave32). Five separable 11-tap Gaussian convs are
// executed as banded 16x16x28 GEMMs on V_WMMA_F32_16X16X4_F32 (7 chained
// K=4 WMMAs each). Everything between the global image load and the scalar
// reduction lives in LDS (~80 KB/WG of the 320 KB WGP pool).
// ---------------------------------------------------------------------------

typedef float v2f __attribute__((ext_vector_type(2)));
typedef float v8f __attribute__((ext_vector_type(8)));

#define KS  11
#define GPAD 5

// f32 WMMA: D(16x16) = A(16x4) * B(4x16) + C
__device__ __forceinline__ v8f wmma4f32(v2f a, v2f b, v8f c) {
  return __builtin_amdgcn_wmma_f32_16x16x4_f32(
      /*neg_a=*/false, a, /*neg_b=*/false, b,
      /*c_mod=*/(short)0, c, /*reuse_a=*/false, /*reuse_b=*/false);
}

// LDS layout (floats), all strides odd to dodge 64-bank conflicts
#define IMG_RS 65
#define IMG_ROWS 64
#define R_S 49
#define MU_S 49
#define P_S 49
#define S_S 33
// pool carve:
//  img1:0  img2:4160                                  (64x65 each)
//  r1:8320 r2:11456   | p11:8320 p22:10672 p12:13024  (r: 64x49, p: 48x49)
//  mu1:15376 mu2:17728                                (48x49 each)
//  s11:0 s22:1584 s12:3168   (alias img space, 48x33 each)
//  red:20080..20208
#define POOL_FLOATS 20208

// Row conv over a 16x16 out tile: out[m0+m][n0+n] = sum_t g[t]*in[m0+m][n0+n+t]
// A = in[m0..m0+15][n0..n0+27] (data), B = banded weights (const regs).
__device__ __forceinline__ void rowconv_tile(const float* in, int si, float* out,
                                             int so, int m0, int n0,
                                             const v2f* w, int lane) {
  const int m  = lane & 15;
  const int kh = lane >> 4;
  v8f acc = {0.f, 0.f, 0.f, 0.f, 0.f, 0.f, 0.f, 0.f};
  const float* p = in + (m0 + m) * si + (n0 + 2 * kh);
#pragma unroll
  for (int k4 = 0; k4 < 7; ++k4) {
    v2f a;
    a.x = p[4 * k4 + 0];
    a.y = p[4 * k4 + 1];
    acc = wmma4f32(a, w[k4], acc);
  }
  const int n = lane & 15;
  float* q = out + n0 + n;
#pragma unroll
  for (int r = 0; r < 8; ++r)
    q[(m0 + r + 8 * kh) * so] = acc[r];
}

// Col conv over a 16x16 out tile: out[m0+m][n0+n] = sum_t g[t]*in[m0+m+t][n0+n]
// A = banded weights (const regs), B = in[m0..m0+27][n0..n0+15] (data).
__device__ __forceinline__ v8f colconv_tile(const float* in, int si, int m0,
                                            int n0, const v2f* w, int lane) {
  const int n  = lane & 15;
  const int kh = lane >> 4;
  v8f acc = {0.f, 0.f, 0.f, 0.f, 0.f, 0.f, 0.f, 0.f};
  const float* p = in + (m0 + 2 * kh) * si + (n0 + n);
#pragma unroll
  for (int k4 = 0; k4 < 7; ++k4) {
    v2f b;
    b.x = p[(4 * k4 + 0) * si];
    b.y = p[(4 * k4 + 1) * si];
    acc = wmma4f32(w[k4], b, acc);
  }
  return acc;
}

__global__ void init_ws_kernel(unsigned* ws) {
  if (threadIdx.x == 0) {
    ws[0] = 0u;           // max accumulator (inputs are >= 0)
    ws[1] = 0x7f7fffffu;  // min accumulator (FLT_MAX bits)
  }
}

__global__ void __launch_bounds__(256)
minmax_kernel(const float* __restrict__ a, const float* __restrict__ b,
              unsigned* ws, int n) {
  __shared__ float shi[256], slo[256];
  const int tid = threadIdx.x;
  float hi = 0.f;
  float lo = __int_as_float(0x7f7fffff);
  for (long i = (long)blockIdx.x * 256 + tid; i < n;
       i += (long)gridDim.x * 256) {
    float x = a[i], y = b[i];
    hi = fmaxf(hi, fmaxf(x, y));
    lo = fminf(lo, fminf(x, y));
  }
  shi[tid] = hi;
  slo[tid] = lo;
  __syncthreads();
  for (int s = 128; s > 0; s >>= 1) {
    if (tid < s) {
      shi[tid] = fmaxf(shi[tid], shi[tid + s]);
      slo[tid] = fminf(slo[tid], slo[tid + s]);
    }
    __syncthreads();
  }
  if (tid == 0) {
    atomicMax(&ws[0], __float_as_uint(shi[0]));  // idempotent -> deterministic
    atomicMin(&ws[1], __float_as_uint(slo[0]));
  }
}

__global__ void __launch_bounds__(128)
ssim_tile_kernel(const float* __restrict__ img1, const float* __restrict__ img2,
                 const unsigned* __restrict__ mmx, float* __restrict__ blockSums,
                 int H, int W) {
  extern __shared__ float pool[];
  float* simg1 = pool;
  float* simg2 = pool + 4160;
  float* r1  = pool + 8320;
  float* r2  = pool + 11456;
  float* p11 = pool + 8320;   // aliases r1/r2 (dead after stage 2)
  float* p22 = pool + 10672;
  float* p12 = pool + 13024;
  float* mu1 = pool + 15376;
  float* mu2 = pool + 17728;
  float* s11 = pool + 0;      // aliases img space (dead after stage 3)
  float* s22 = pool + 1584;
  float* s12 = pool + 3168;
  float* red = pool + 20080;

  const int tid  = threadIdx.x;
  const int lane = tid & 31;
  const int wave = tid >> 5;

  const int tcol  = blockIdx.x * 32;
  const int trow  = blockIdx.y * 32;
  const long imgOff = (long)blockIdx.z * H * W;
  const float* I1 = img1 + imgOff;
  const float* I2 = img2 + imgOff;

  const float vr = __uint_as_float(mmx[0]) - __uint_as_float(mmx[1]) + 1e-5f;
  const float C1 = (0.01f * vr) * (0.01f * vr);
  const float C2 = (0.03f * vr) * (0.03f * vr);

  // Gaussian taps (sigma = 20) and banded weight registers.
  // Both operand roles need W[k][j] = g[k-j] with j = lane&15, k = 4*k4+2*kh{,+1},
  // so a single per-lane v2f[7] serves as row-conv B and col-conv A.
  float g[KS];
  {
    float s = 0.f;
#pragma unroll
    for (int t = 0; t < KS; ++t) {
      float d = (float)(t - GPAD);
      g[t] = expf(-d * d * (1.f / 800.f));
      s += g[t];
    }
    float inv = 1.f / s;
#pragma unroll
    for (int t = 0; t < KS; ++t) g[t] *= inv;
  }
  v2f w[7];
  {
    const int j  = lane & 15;
    const int kh = lane >> 4;
#pragma unroll
    for (int k4 = 0; k4 < 7; ++k4) {
      int k0 = 4 * k4 + 2 * kh;
      int d0 = k0 - j, d1 = k0 + 1 - j;
      w[k4].x = (d0 >= 0 && d0 < KS) ? g[d0] : 0.f;
      w[k4].y = (d1 >= 0 && d1 < KS) ? g[d1] : 0.f;
    }
  }

  // Stage 0: load both 52x52 halo tiles (zero-padded) into 64x65 LDS frames.
  const int r0 = trow - 10, c0 = tcol - 10;
  for (int i = tid; i < IMG_ROWS * IMG_RS; i += 128) {
    int lr = i / IMG_RS, lc = i % IMG_RS;
    int gr = r0 + lr, gc = c0 + lc;
    bool ok = (lr < 52) && (lc < 52) && (gr >= 0) && (gr < H) &&
              (gc >= 0) && (gc < W);
    long off = (long)gr * W + gc;
    simg1[i] = ok ? I1[off] : 0.f;
    simg2[i] = ok ? I2[off] : 0.f;
  }
  __syncthreads();

  // Stage 1: row conv of both images -> r (64 rows x 48 cols), 24 tile-ops.
  for (int o = wave; o < 24; o += 4) {
    int im = o & 1, t = o >> 1;
    int m0 = (t / 3) * 16, n0 = (t % 3) * 16;
    rowconv_tile(im ? simg2 : simg1, IMG_RS, im ? r2 : r1, R_S, m0, n0, w, lane);
  }
  __syncthreads();

  // Stage 2: col conv -> mu (48x48), 18 tile-ops.
  for (int o = wave; o < 18; o += 4) {
    int im = o & 1, t = o >> 1;
    int m0 = (t / 3) * 16, n0 = (t % 3) * 16;
    v8f acc = colconv_tile(im ? r2 : r1, R_S, m0, n0, w, lane);
    float* mu = im ? mu2 : mu1;
    const int n = lane & 15, kh = lane >> 4;
#pragma unroll
    for (int r = 0; r < 8; ++r)
      mu[(m0 + r + 8 * kh) * MU_S + n0 + n] = acc[r];
  }
  __syncthreads();

  // Stage 3: residual products, masked to zero outside the image
  // (reference zero-pads d1*d1, not (0-mu)^2).
  for (int i = tid; i < 48 * 48; i += 128) {
    int r = i / 48, c = i % 48;
    int gr = trow - 5 + r, gc = tcol - 5 + c;
    bool in_img = (gr >= 0) && (gr < H) && (gc >= 0) && (gc < W);
    float a1 = simg1[(r + 5) * IMG_RS + (c + 5)] - mu1[r * MU_S + c];
    float a2 = simg2[(r + 5) * IMG_RS + (c + 5)] - mu2[r * MU_S + c];
    if (!in_img) { a1 = 0.f; a2 = 0.f; }
    p11[r * P_S + c] = a1 * a1;
    p22[r * P_S + c] = a2 * a2;
    p12[r * P_S + c] = a1 * a2;
  }
  __syncthreads();

  // Stage 4: row conv of products -> s (48 rows x 32 cols), 18 tile-ops.
  for (int o = wave; o < 18; o += 4) {
    int pr = o % 3, t = o / 3;
    int m0 = (t / 2) * 16, n0 = (t % 2) * 16;
    const float* pin = (pr == 0) ? p11 : ((pr == 1) ? p22 : p12);
    float* sout      = (pr == 0) ? s11 : ((pr == 1) ? s22 : s12);
    rowconv_tile(pin, P_S, sout, S_S, m0, n0, w, lane);
  }
  __syncthreads();

  // Stage 5: col conv of s + SSIM combine; one 16x16 out tile per wave.
  {
    const int mt = (wave >> 1) * 16;
    const int nt = (wave & 1) * 16;
    v8f o11 = colconv_tile(s11, S_S, mt, nt, w, lane);
    v8f o22 = colconv_tile(s22, S_S, mt, nt, w, lane);
    v8f o12 = colconv_tile(s12, S_S, mt, nt, w, lane);
    const int n = lane & 15, kh = lane >> 4;
    float sum = 0.f;
#pragma unroll
    for (int r = 0; r < 8; ++r) {
      int orow = mt + r + 8 * kh;
      int ocol = nt + n;
      float m1 = mu1[(orow + 5) * MU_S + (ocol + 5)];
      float m2 = mu2[(orow + 5) * MU_S + (ocol + 5)];
      float sg1  = o11[r] + 1.f;
      float sg2  = o22[r] + 1.f;
      float sg12 = o12[r] + 1.f;
      float m1s = m1 * m1 + 1.f;
      float m2s = m2 * m2 + 1.f;
      float m12 = 2.f * (m1 * m2 + 1.f);
      float structure = (2.f * sg12 + C2) / (sg1 + sg2 + C2);
      float bright = (m12 * m12 + C1) / ((m1s + m2s) * (m1s + m2s) + C1);
      sum += structure * bright;
    }
    red[tid] = sum;
  }
  __syncthreads();
  for (int s = 64; s > 0; s >>= 1) {
    if (tid < s) red[tid] += red[tid + s];
    __syncthreads();
  }
  if (tid == 0) {
    int bid = (blockIdx.z * gridDim.y + blockIdx.y) * gridDim.x + blockIdx.x;
    blockSums[bid] = red[0];
  }
}

__global__ void __launch_bounds__(256)
finalize_kernel(const float* __restrict__ bsum, int nb, float* out, float invN) {
  __shared__ float red[256];
  const int tid = threadIdx.x;
  float s = 0.f;
  for (int i = tid; i < nb; i += 256) s += bsum[i];  // fixed order
  red[tid] = s;
  __syncthreads();
  for (int k = 128; k > 0; k >>= 1) {
    if (tid < k) red[tid] += red[tid + k];
    __syncthreads();
  }
  if (tid == 0) out[0] = 1.f - red[0] * invN;
}

extern "C" void kernel_launch(void* const* d_in, const int* in_sizes, int n_in,
                              void* d_out, int out_size, void* d_ws,
                              size_t ws_size, hipStream_t stream) {
  (void)in_sizes; (void)n_in; (void)out_size; (void)ws_size;
  const float* img1 = (const float*)d_in[0];
  const float* img2 = (const float*)d_in[1];
  const int B = 16, H = 1024, W = 1024;
  const int n = B * H * W;

  unsigned* wsu = (unsigned*)d_ws;
  float* blockSums = (float*)d_ws + 16;
  const int tilesX = W / 32, tilesY = H / 32;
  const int nb = B * tilesX * tilesY;

  init_ws_kernel<<<1, 64, 0, stream>>>(wsu);
  minmax_kernel<<<2048, 256, 0, stream>>>(img1, img2, wsu, n);

  dim3 grid(tilesX, tilesY, B);
  size_t smem = (size_t)POOL_FLOATS * sizeof(float);
  ssim_tile_kernel<<<grid, 128, smem, stream>>>(img1, img2, wsu, blockSums, H, W);

  finalize_kernel<<<1, 256, 0, stream>>>(blockSums, nb, (float*)d_out,
                                         1.f / (float)n);
}